// BatchDelayProcessor_73160472920638
// MI455X (gfx1250) — compile-verified
//
#include <hip/hip_runtime.h>
#include <cstdint>

// Problem constants (from the reference)
#define SR_T   441000          // samples per batch row
#define SR_D   22050           // delay length = block length
#define NBLK   20              // SR_T / SR_D
#define NB     64              // batch
#define HALF   (SR_D / 2)      // 11025 float2-chains per batch row
#define TPB    256

// Issue an async global->LDS 8-byte copy. The preceding s_wait_dscnt 0
// guarantees every earlier ds_load has delivered its data to VGPRs before the
// async engine may overwrite the staging buffer (DS reads and async-LDS
// writes are not mutually ordered).
__device__ __forceinline__ void async_stage_b64_guarded(uint32_t lds_addr, const void* gaddr) {
  asm volatile("s_wait_dscnt 0x0\n\t"
               "global_load_async_to_lds_b64 %0, %1, off"
               :: "v"(lds_addr), "v"(gaddr)
               : "memory");
}

__device__ __forceinline__ void async_stage_b64(uint32_t lds_addr, const void* gaddr) {
  asm volatile("global_load_async_to_lds_b64 %0, %1, off"
               :: "v"(lds_addr), "v"(gaddr)
               : "memory");
}

__global__ __launch_bounds__(TPB) void delay_fb_kernel(const float* __restrict__ xg,
                                                       float* __restrict__ og) {
  __shared__ float2 stage[2][TPB];   // double-buffered staging: 4 KB per workgroup

  const int tid    = threadIdx.x;
  const int jj_raw = blockIdx.x * TPB + tid;      // float2-chain index within row
  const int jj     = (jj_raw < HALF) ? jj_raw : (HALF - 1);  // clamp tail lanes
  const bool do_store = (jj_raw < HALF);
  const int b      = blockIdx.y;

  const float2* __restrict__ xrow =
      reinterpret_cast<const float2*>(xg) + (size_t)b * (SR_T / 2) + jj;
  float2* __restrict__ orow =
      reinterpret_cast<float2*>(og) + (size_t)b * (SR_T / 2) + jj;

  // LDS byte addresses for this lane's two staging slots.
  // Generic shared pointers carry the LDS byte offset in their low 32 bits
  // (ISA 10.2).
  uint32_t lds_addr[2];
  lds_addr[0] = (uint32_t)(uintptr_t)&stage[0][tid];
  lds_addr[1] = (uint32_t)(uintptr_t)&stage[1][tid];

  // Prime the pipeline: block 0 -> buffer 0.
  async_stage_b64(lds_addr[0], (const void*)xrow);

  float cx = 0.0f, cy = 0.0f;   // feedback carry (delay-line state), one per chain
#pragma unroll
  for (int k = 0; k < NBLK; ++k) {
    if (k + 1 < NBLK) {
      // Prefetch block k+1 into the other buffer, then retire block k's copy.
      // Two async loads may be in flight; they complete in order, so
      // asynccnt <= 1 means the older (block k) has landed in LDS.
      async_stage_b64_guarded(lds_addr[(k + 1) & 1],
                              (const void*)(xrow + (size_t)(k + 1) * HALF));
      asm volatile("s_wait_asynccnt 0x1" ::: "memory");
    } else {
      asm volatile("s_wait_asynccnt 0x0" ::: "memory");
    }

    float2 v = stage[k & 1][tid];        // ds_load_b64 of the staged tile

    float2 o;
    o.x = v.x * 0.5f + cx * 0.5f;        // out = x*(1-MIX) + delayed*MIX, MIX=0.5
    o.y = v.y * 0.5f + cy * 0.5f;
    cx  = fmaf(0.3f, cx, v.x);           // carry = x + FEEDBACK*delayed
    cy  = fmaf(0.3f, cy, v.y);

    if (do_store) orow[(size_t)k * HALF] = o;   // global_store_b64, coalesced
  }
}

extern "C" void kernel_launch(void* const* d_in, const int* in_sizes, int n_in,
                              void* d_out, int out_size, void* d_ws, size_t ws_size,
                              hipStream_t stream) {
  (void)in_sizes; (void)n_in; (void)d_ws; (void)ws_size; (void)out_size;
  const float* x = (const float*)d_in[0];
  float* out     = (float*)d_out;

  dim3 grid((HALF + TPB - 1) / TPB, NB, 1);   // 44 x 64 workgroups
  delay_fb_kernel<<<grid, dim3(TPB, 1, 1), 0, stream>>>(x, out);
}